// Encoder_45956150067649
// MI455X (gfx1250) — compile-verified
//
#include <hip/hip_runtime.h>
#include <hip/hip_bf16.h>

// ---------------- problem constants (from reference) ----------------
constexpr int N       = 30000;
constexpr int E       = 960000;
constexpr int IN_DIM  = 2048;
constexpr int N_EMBED = 64;
constexpr int N_HID1  = 32;
constexpr int N_HID2  = 16;
constexpr int NUM_ET  = 64;
constexpr int NUM_BASE= 16;

typedef float v2f __attribute__((ext_vector_type(2)));
typedef float v8f __attribute__((ext_vector_type(8)));

// D = A(16x4 f32) * B(4x16 f32) + C(16x16 f32), wave32, exact fp32
__device__ __forceinline__ v8f wmma4(v2f a, v2f b, v8f c) {
    return __builtin_amdgcn_wmma_f32_16x16x4_f32(
        /*neg_a=*/false, a, /*neg_b=*/false, b,
        /*c_mod=*/(short)0, c, /*reuse_a=*/false, /*reuse_b=*/false);
}

// Async DMA: 16B global -> LDS, tracked by ASYNCcnt (no VGPR round-trip).
// lds = 32-bit LDS byte offset (per-lane), gptr = 64-bit global address (per-lane).
__device__ __forceinline__ void async_g2lds_b128(uint32_t lds, const void* gptr) {
    asm volatile("global_load_async_to_lds_b128 %0, %1, off"
                 :: "v"(lds), "v"(gptr) : "memory");
}
__device__ __forceinline__ void wait_asynccnt0() {
    asm volatile("s_wait_asynccnt 0" ::: "memory");
}

// ---------------- W[r] = sum_b att[r,b] * basis[b]  ----------------
__global__ void make_w_kernel(const float* __restrict__ att,
                              const float* __restrict__ basis,
                              float* __restrict__ W, int sz /* IN*OUT */) {
    const int r = blockIdx.x;
    const float* a = att + r * NUM_BASE;
    for (int i = threadIdx.x; i < sz; i += blockDim.x) {
        float s = 0.f;
#pragma unroll
        for (int b = 0; b < NUM_BASE; ++b)
            s = fmaf(a[b], basis[(size_t)b * sz + i], s);
        W[(size_t)r * sz + i] = s;
    }
}

// ---------- h0 = (x @ embed) / x_norm : 30000x2048x64 WMMA GEMM ----------
// block = 128 threads = 4 waves; block computes 16 rows x 64 cols; wave w owns cols [16w,16w+16)
// Tiles staged via GLOBAL_LOAD_ASYNC_TO_LDS_B128 (ASYNCcnt), consumed by WMMA from LDS.
__global__ __launch_bounds__(128) void embed_kernel(const float* __restrict__ x,
                                                    const float* __restrict__ emb,
                                                    const float* __restrict__ x_norm,
                                                    float* __restrict__ h0) {
    // pitches chosen so every 16B async write is 16B-aligned in LDS
    // (132*4 = 528 = 33*16 ; 68*4 = 272 = 17*16) and fragment reads stay bank-conflict-free.
    __shared__ float xs[16][132];     // 16 rows x 128 K-chunk
    __shared__ float es[128][68];     // 128 K x 64 cols
    const int tid  = threadIdx.x;
    const int wave = tid >> 5;
    const int lane = tid & 31;
    const int half = lane >> 4;
    const int m    = lane & 15;
    const int r0   = blockIdx.x * 16;
    const int c0   = wave * 16;

    const uint32_t xs_base = (uint32_t)(uintptr_t)&xs[0][0];
    const uint32_t es_base = (uint32_t)(uintptr_t)&es[0][0];

    v8f acc = {};
    for (int kb = 0; kb < IN_DIM; kb += 128) {
        // ---- async-stage x tile: 2048 floats = 4 x B128 per thread ----
#pragma unroll
        for (int i = 0; i < 4; ++i) {
            const int idx = tid * 4 + i * 512;          // float index in 16x128 tile
            const int row = idx >> 7, c = idx & 127;
            async_g2lds_b128(xs_base + (uint32_t)(row * 132 + c) * 4u,
                             &x[(size_t)(r0 + row) * IN_DIM + kb + c]);
        }
        // ---- async-stage embed tile: 8192 floats = 16 x B128 per thread ----
#pragma unroll
        for (int i = 0; i < 16; ++i) {
            const int idx = tid * 4 + i * 512;          // float index in 128x64 tile
            const int row = idx >> 6, c = idx & 63;
            async_g2lds_b128(es_base + (uint32_t)(row * 68 + c) * 4u,
                             &emb[(size_t)(kb + row) * 64 + c]);
        }
        // hint the next streaming x chunk toward the caches while DMA runs
        if (kb + 128 < IN_DIM)
            __builtin_prefetch(&x[(size_t)(r0 + (tid & 15)) * IN_DIM + kb + 128 + ((tid >> 4) << 4)], 0, 0);
        wait_asynccnt0();
        __syncthreads();
#pragma unroll
        for (int k = 0; k < 128; k += 4) {
            v2f a, b;
            a.x = xs[m][k + 2 * half];
            a.y = xs[m][k + 2 * half + 1];
            b.x = es[k + 2 * half][c0 + m];
            b.y = es[k + 2 * half + 1][c0 + m];
            acc = wmma4(a, b, acc);
        }
        __syncthreads();
    }
#pragma unroll
    for (int j = 0; j < 8; ++j) {
        const int row = r0 + j + 8 * half;   // D layout: VGPR j -> rows j / j+8
        h0[(size_t)row * N_EMBED + c0 + m] = acc[j] / x_norm[row];
    }
}

// ---------- per-edge message + scatter-add, relation-segmented WMMA ----------
// block = 128 threads = 4 waves; each wave handles 16 consecutive edges.
// edge_type is sorted, so 16-edge groups are relation-uniform except at boundaries.
template <int KD, int OD>
__global__ __launch_bounds__(128) void edge_msg_kernel(const float* __restrict__ h,
                                                       const float* __restrict__ W,
                                                       const int* __restrict__ src,
                                                       const int* __restrict__ dst,
                                                       const int* __restrict__ et,
                                                       float* __restrict__ agg) {
    __shared__ int   sSrc[4][16];
    __shared__ int   sDst[4][16];
    __shared__ float sA[4][16][KD + 1];   // gathered source rows (coalesced stage)

    const int tid  = threadIdx.x;
    const int wave = tid >> 5;
    const int lane = tid & 31;
    const int half = lane >> 4;
    const int m    = lane & 15;
    const int e0   = blockIdx.x * 64 + wave * 16;
    if (e0 >= E) return;

    const int nvalid = min(16, E - e0);
    if (lane < 16) {
        const int ee = min(e0 + lane, E - 1);
        sSrc[wave][lane] = src[ee];
        sDst[wave][lane] = dst[ee];
    }
    const int et0 = et[e0];
    const bool uniform = (nvalid == 16) && (et[e0 + 15] == et0);

    if (uniform) {
        const float* Wr = W + (size_t)et0 * KD * OD;
        // coalesced gather of the 16 source rows into LDS (in-order per wave)
        for (int idx = lane; idx < 16 * KD; idx += 32) {
            const int row = idx / KD, c = idx - row * KD;
            sA[wave][row][c] = h[(size_t)sSrc[wave][row] * KD + c];
        }
#pragma unroll
        for (int nt = 0; nt < OD / 16; ++nt) {
            v8f acc = {};
#pragma unroll
            for (int k = 0; k < KD; k += 4) {
                v2f a, b;
                a.x = sA[wave][m][k + 2 * half];
                a.y = sA[wave][m][k + 2 * half + 1];
                b.x = Wr[(size_t)(k + 2 * half) * OD + nt * 16 + m];
                b.y = Wr[(size_t)(k + 2 * half + 1) * OD + nt * 16 + m];
                acc = wmma4(a, b, acc);
            }
#pragma unroll
            for (int j = 0; j < 8; ++j) {
                const int erow = j + 8 * half;             // edge within group
                const int d    = sDst[wave][erow];
                atomicAdd(&agg[(size_t)d * OD + nt * 16 + m], acc[j]);
            }
        }
    } else {
        // boundary / tail group: per-lane scalar matvec (rare)
        if (lane < nvalid) {
            const int ee = e0 + lane;
            const int s = src[ee], d = dst[ee];
            const float* Wr = W + (size_t)et[ee] * KD * OD;
            for (int o = 0; o < OD; ++o) {
                float sum = 0.f;
#pragma unroll
                for (int i = 0; i < KD; ++i)
                    sum = fmaf(h[(size_t)s * KD + i], Wr[(size_t)i * OD + o], sum);
                atomicAdd(&agg[(size_t)d * OD + o], sum);
            }
        }
    }
}

// ---------- hout = relu(agg + hin @ root + bias), WMMA root GEMM ----------
// block = 32*(OD/16) threads; block computes 16 rows x OD cols
template <int KD, int OD>
__global__ __launch_bounds__(64) void root_kernel(const float* __restrict__ hin,
                                                  const float* __restrict__ agg,
                                                  const float* __restrict__ root,
                                                  const float* __restrict__ bias,
                                                  float* __restrict__ hout) {
    __shared__ float hs[16][KD + 1];
    __shared__ float rs[KD][OD];
    const int tid  = threadIdx.x;
    const int wave = tid >> 5;
    const int lane = tid & 31;
    const int half = lane >> 4;
    const int m    = lane & 15;
    const int r0   = blockIdx.x * 16;
    const int c0   = wave * 16;
    const int nthr = blockDim.x;

    for (int i = tid; i < 16 * KD; i += nthr)
        hs[i / KD][i % KD] = hin[(size_t)(r0 + i / KD) * KD + (i % KD)];
    for (int i = tid; i < KD * OD; i += nthr)
        rs[i / OD][i % OD] = root[i];
    __syncthreads();

    v8f acc = {};
#pragma unroll
    for (int k = 0; k < KD; k += 4) {
        v2f a, b;
        a.x = hs[m][k + 2 * half];
        a.y = hs[m][k + 2 * half + 1];
        b.x = rs[k + 2 * half][c0 + m];
        b.y = rs[k + 2 * half + 1][c0 + m];
        acc = wmma4(a, b, acc);
    }
#pragma unroll
    for (int j = 0; j < 8; ++j) {
        const int row = r0 + j + 8 * half;
        const int col = c0 + m;
        float v = acc[j] + agg[(size_t)row * OD + col] + bias[col];
        hout[(size_t)row * OD + col] = v > 0.f ? v : 0.f;
    }
}

// ------------------------------- launcher -------------------------------
extern "C" void kernel_launch(void* const* d_in, const int* in_sizes, int n_in,
                              void* d_out, int out_size, void* d_ws, size_t ws_size,
                              hipStream_t stream) {
    (void)in_sizes; (void)n_in; (void)out_size; (void)ws_size;
    const float* x        = (const float*)d_in[0];
    const int*   eidx     = (const int*)  d_in[1];
    const int*   etype    = (const int*)  d_in[2];
    /* d_in[3] = range_list (unused; derived from sorted edge_type) */
    const float* x_norm   = (const float*)d_in[4];
    const float* emb      = (const float*)d_in[5];
    const float* basis1   = (const float*)d_in[6];
    const float* att1     = (const float*)d_in[7];
    const float* root1    = (const float*)d_in[8];
    const float* bias1    = (const float*)d_in[9];
    const float* basis2   = (const float*)d_in[10];
    const float* att2     = (const float*)d_in[11];
    const float* root2    = (const float*)d_in[12];
    const float* bias2    = (const float*)d_in[13];
    float* out = (float*)d_out;

    // workspace carve-up (~18 MB)
    float* h0   = (float*)d_ws;                 // [N, 64]
    float* h1   = h0   + (size_t)N * N_EMBED;   // [N, 32]
    float* agg1 = h1   + (size_t)N * N_HID1;    // [N, 32]
    float* agg2 = agg1 + (size_t)N * N_HID1;    // [N, 16]
    float* W1   = agg2 + (size_t)N * N_HID2;    // [64, 64, 32]
    float* W2   = W1   + (size_t)NUM_ET * N_EMBED * N_HID1; // [64, 32, 16]

    const int* src = eidx;
    const int* dst = eidx + E;

    make_w_kernel<<<NUM_ET, 256, 0, stream>>>(att1, basis1, W1, N_EMBED * N_HID1);
    make_w_kernel<<<NUM_ET, 256, 0, stream>>>(att2, basis2, W2, N_HID1 * N_HID2);

    embed_kernel<<<N / 16, 128, 0, stream>>>(x, emb, x_norm, h0);

    hipMemsetAsync(agg1, 0, (size_t)N * N_HID1 * sizeof(float), stream);
    edge_msg_kernel<N_EMBED, N_HID1><<<(E + 63) / 64, 128, 0, stream>>>(h0, W1, src, dst, etype, agg1);
    root_kernel<N_EMBED, N_HID1><<<N / 16, 64, 0, stream>>>(h0, agg1, root1, bias1, h1);

    hipMemsetAsync(agg2, 0, (size_t)N * N_HID2 * sizeof(float), stream);
    edge_msg_kernel<N_HID1, N_HID2><<<(E + 63) / 64, 128, 0, stream>>>(h1, W2, src, dst, etype, agg2);
    root_kernel<N_HID1, N_HID2><<<N / 16, 32, 0, stream>>>(h1, agg2, root2, bias2, out);
}